// GCN3_bias_2adv_20727512170665
// MI455X (gfx1250) — compile-verified
//
#include <hip/hip_runtime.h>
#include <hip/hip_bf16.h>

#define U_NUM 359347
#define I_NUM 292589
#define DF    64
#define NNZ_N 4000000
#define B_N   32768
#define LAM   0.001f

typedef __attribute__((ext_vector_type(2))) float v2f;
typedef __attribute__((ext_vector_type(8))) float v8f;

// ---------------------------------------------------------------------------
// SpMM (both directions fused): one wave per edge, each lane handles 2 dims.
// acc_u[r,:] += vu * x_item[c,:]   ;   acc_i[c,:] += vi * x_user[r,:]
// Tables fit in the 192MB L2, so these atomics/gathers stay on-chip.
// ---------------------------------------------------------------------------
__global__ void spmm_both_kernel(const int* __restrict__ rows,
                                 const int* __restrict__ cols,
                                 const float* __restrict__ vu,
                                 const float* __restrict__ vi,
                                 const float* __restrict__ x_item,
                                 const float* __restrict__ x_user,
                                 float* __restrict__ acc_u,
                                 float* __restrict__ acc_i,
                                 int nnz) {
    int wave = (int)((blockIdx.x * (size_t)blockDim.x + threadIdx.x) >> 5);
    int lane = threadIdx.x & 31;
    if (wave >= nnz) return;
    int r = rows[wave];
    int c = cols[wave];
    float vuu = vu[wave];
    float vii = vi[wave];
    int d = lane * 2;
    float2 gi = *(const float2*)(x_item + (size_t)c * DF + d);
    atomicAdd(acc_u + (size_t)r * DF + d,     vuu * gi.x);
    atomicAdd(acc_u + (size_t)r * DF + d + 1, vuu * gi.y);
    float2 gu = *(const float2*)(x_user + (size_t)r * DF + d);
    atomicAdd(acc_i + (size_t)c * DF + d,     vii * gu.x);
    atomicAdd(acc_i + (size_t)c * DF + d + 1, vii * gu.y);
}

// g[i] = relu(g[i] + base[i] * deg[i/64])   (in-place on the SpMM accumulator)
__global__ void resrelu_kernel(float* __restrict__ g,
                               const float* __restrict__ base,
                               const float* __restrict__ deg,
                               size_t n) {
    size_t i = blockIdx.x * (size_t)blockDim.x + threadIdx.x;
    if (i >= n) return;
    float v = g[i] + base[i] * deg[i >> 6];
    g[i] = v > 0.0f ? v : 0.0f;
}

// sum over n of (w0*e0 + w1*g1 + w2*g2)^2, accumulated into *out (float atomic)
__global__ void ssq_kernel(const float* __restrict__ e0,
                           const float* __restrict__ g1,
                           const float* __restrict__ g2,
                           const float* __restrict__ add_w,
                           float* __restrict__ out,
                           size_t n) {
    float w0 = add_w[0], w1 = add_w[1], w2 = add_w[2];
    float acc = 0.0f;
    size_t stride = (size_t)gridDim.x * blockDim.x;
    for (size_t i = blockIdx.x * (size_t)blockDim.x + threadIdx.x; i < n; i += stride) {
        float v = w0 * e0[i] + w1 * g1[i] + w2 * g2[i];
        acc += v * v;
    }
    // wave32 reduction
    for (int off = 16; off > 0; off >>= 1)
        acc += __shfl_down(acc, off, 32);
    if ((threadIdx.x & 31) == 0)
        atomicAdd(out, acc);
}

// out[b,d] = w0*e0[idx[b],d] + w1*g1[idx[b],d] + w2*g2[idx[b],d]
__global__ void gather_comb_kernel(const int* __restrict__ idx,
                                   const float* __restrict__ e0,
                                   const float* __restrict__ g1,
                                   const float* __restrict__ g2,
                                   const float* __restrict__ add_w,
                                   float* __restrict__ out,
                                   int bn) {
    int t = (int)(blockIdx.x * (size_t)blockDim.x + threadIdx.x);
    if (t >= bn * DF) return;
    int b = t >> 6;
    int d = t & 63;
    size_t ro = (size_t)idx[b] * DF + d;
    out[t] = add_w[0] * e0[ro] + add_w[1] * g1[ro] + add_w[2] * g2[ro];
}

// ---------------------------------------------------------------------------
// Dense GEMM via V_WMMA_F32_16X16X4_F32 (exact fp32):
//   out[m,n] = leaky( sum_k A[m,k] * W[n,k] + bias[n] )
// A: [M,K] row-major, W: [N,K] row-major (i.e. out = A @ W^T + bias).
// One wave per 16x16 tile; M,N,K all multiples of 16 -> EXEC stays all-ones.
// ---------------------------------------------------------------------------
__global__ void gemm_leaky_wmma(const float* __restrict__ A,
                                const float* __restrict__ W,
                                const float* __restrict__ bias,
                                float* __restrict__ out,
                                int M, int N, int K) {
    int wave = (int)((blockIdx.x * (size_t)blockDim.x + threadIdx.x) >> 5);
    int ntn = N >> 4;
    int tm = wave / ntn;
    int tn = wave - tm * ntn;
    if (tm >= (M >> 4)) return;   // uniform per wave
    int lane = threadIdx.x & 31;
    int half = lane >> 4;         // 0: K pair {0,1}, 1: K pair {2,3}
    int l    = lane & 15;

    const float* arow = A + (size_t)(tm * 16 + l) * K;   // A[M=l of tile][.]
    const float* wrow = W + (size_t)(tn * 16 + l) * K;   // B[k][n=l] = W[l][k]

    v8f acc = {};
    for (int kk = 0; kk < K; kk += 4) {
        v2f a, b;
        a.x = arow[kk + 2 * half];
        a.y = arow[kk + 2 * half + 1];
        b.x = wrow[kk + 2 * half];
        b.y = wrow[kk + 2 * half + 1];
        acc = __builtin_amdgcn_wmma_f32_16x16x4_f32(
            /*neg_a=*/false, a, /*neg_b=*/false, b,
            /*c_mod=*/(short)0, acc, /*reuse_a=*/false, /*reuse_b=*/false);
    }

    float bv  = bias[tn * 16 + l];
    int   col = tn * 16 + l;
    int   mb  = tm * 16 + half * 8;
#pragma unroll
    for (int v = 0; v < 8; ++v) {
        float val = acc[v] + bv;
        val = val > 0.0f ? val : 0.1f * val;   // leaky relu
        out[(size_t)(mb + v) * N + col] = val;
    }
}

// one wave per batch element: dot(user,item) + biases + avg; accumulate (pred-r)^2
__global__ void pred_loss_kernel(const float* __restrict__ user,
                                 const float* __restrict__ item,
                                 const int* __restrict__ u0,
                                 const int* __restrict__ i0,
                                 const float* __restrict__ ub,
                                 const float* __restrict__ ib,
                                 const float* __restrict__ avg,
                                 const float* __restrict__ ratings,
                                 float* __restrict__ acc,
                                 int bn) {
    int wave = (int)((blockIdx.x * (size_t)blockDim.x + threadIdx.x) >> 5);
    int lane = threadIdx.x & 31;
    if (wave >= bn) return;
    const float* ur = user + (size_t)wave * DF;
    const float* ir = item + (size_t)wave * DF;
    float s = ur[lane] * ir[lane] + ur[lane + 32] * ir[lane + 32];
    for (int off = 16; off > 0; off >>= 1)
        s += __shfl_down(s, off, 32);
    if (lane == 0) {
        float pred = s + ub[u0[wave]] + ib[i0[wave]] + avg[0];
        float d = pred - ratings[wave];
        atomicAdd(acc, d * d);
    }
}

__global__ void finalize_kernel(const float* __restrict__ scal, float* __restrict__ out) {
    if (threadIdx.x == 0 && blockIdx.x == 0) {
        float mse   = scal[0] / (float)B_N;
        float ssq_u = scal[1] / ((float)U_NUM * (float)DF);
        float ssq_i = scal[2] / ((float)I_NUM * (float)DF);
        out[0] = mse + LAM * ssq_u + LAM * ssq_i;
    }
}

extern "C" void kernel_launch(void* const* d_in, const int* in_sizes, int n_in,
                              void* d_out, int out_size, void* d_ws, size_t ws_size,
                              hipStream_t stream) {
    const int*   ui_rows   = (const int*)  d_in[0];
    const int*   ui_cols   = (const int*)  d_in[1];
    const float* ui_vals   = (const float*)d_in[2];
    const float* iu_vals   = (const float*)d_in[3];
    const float* d_i       = (const float*)d_in[4];
    const float* d_j       = (const float*)d_in[5];
    const float* ue        = (const float*)d_in[6];
    const float* ie        = (const float*)d_in[7];
    const float* add_w     = (const float*)d_in[8];
    const float* fw1       = (const float*)d_in[9];   // [128,64]
    const float* fb1       = (const float*)d_in[10];  // [128]
    const float* fw2       = (const float*)d_in[11];  // [64,128]
    const float* fb2       = (const float*)d_in[12];  // [64]
    const float* user_bias = (const float*)d_in[13];
    const float* item_bias = (const float*)d_in[14];
    const float* avg       = (const float*)d_in[15];
    const int*   user0     = (const int*)  d_in[16];
    const int*   item_i0   = (const int*)  d_in[17];
    const float* ratings   = (const float*)d_in[18];
    float*       out       = (float*)d_out;

    // -------- workspace layout (floats) --------
    float* ws = (float*)d_ws;
    size_t off = 0;
    float* g1_u = ws + off; off += (size_t)U_NUM * DF;
    float* g1_i = ws + off; off += (size_t)I_NUM * DF;
    float* g2_u = ws + off; off += (size_t)U_NUM * DF;
    float* g2_i = ws + off; off += (size_t)I_NUM * DF;
    size_t acc_floats = off;                 // contiguous accumulator region
    float* xu   = ws + off; off += (size_t)B_N * DF;
    float* xi   = ws + off; off += (size_t)B_N * DF;
    float* hu   = ws + off; off += (size_t)B_N * 2 * DF;
    float* hi   = ws + off; off += (size_t)B_N * 2 * DF;
    float* fu   = ws + off; off += (size_t)B_N * DF;
    float* fi   = ws + off; off += (size_t)B_N * DF;
    float* scal = ws + off; off += 4;        // [mse_sum, ssq_u, ssq_i]

    // -------- zero accumulators + scalars --------
    hipMemsetAsync(ws,   0, acc_floats * sizeof(float), stream);
    hipMemsetAsync(scal, 0, 4 * sizeof(float), stream);

    const int TB = 256;

    // -------- layer 1 SpMM (both directions) --------
    {
        int waves_per_block = TB / 32;
        int blocks = (NNZ_N + waves_per_block - 1) / waves_per_block;
        spmm_both_kernel<<<blocks, TB, 0, stream>>>(ui_rows, ui_cols, ui_vals, iu_vals,
                                                    ie, ue, g1_u, g1_i, NNZ_N);
    }
    {
        size_t n = (size_t)U_NUM * DF;
        resrelu_kernel<<<(unsigned)((n + TB - 1) / TB), TB, 0, stream>>>(g1_u, ue, d_i, n);
        n = (size_t)I_NUM * DF;
        resrelu_kernel<<<(unsigned)((n + TB - 1) / TB), TB, 0, stream>>>(g1_i, ie, d_j, n);
    }

    // -------- layer 2 SpMM (inputs are post-relu layer-1 outputs) --------
    {
        int waves_per_block = TB / 32;
        int blocks = (NNZ_N + waves_per_block - 1) / waves_per_block;
        spmm_both_kernel<<<blocks, TB, 0, stream>>>(ui_rows, ui_cols, ui_vals, iu_vals,
                                                    g1_i, g1_u, g2_u, g2_i, NNZ_N);
    }
    {
        size_t n = (size_t)U_NUM * DF;
        resrelu_kernel<<<(unsigned)((n + TB - 1) / TB), TB, 0, stream>>>(g2_u, g1_u, d_i, n);
        n = (size_t)I_NUM * DF;
        resrelu_kernel<<<(unsigned)((n + TB - 1) / TB), TB, 0, stream>>>(g2_i, g1_i, d_j, n);
    }

    // -------- L2 regularization sums (gcn computed on the fly) --------
    ssq_kernel<<<4096, TB, 0, stream>>>(ue, g1_u, g2_u, add_w, scal + 1, (size_t)U_NUM * DF);
    ssq_kernel<<<4096, TB, 0, stream>>>(ie, g1_i, g2_i, add_w, scal + 2, (size_t)I_NUM * DF);

    // -------- batch gather + 3-layer combination --------
    {
        int n = B_N * DF;
        gather_comb_kernel<<<(n + TB - 1) / TB, TB, 0, stream>>>(user0, ue, g1_u, g2_u, add_w, xu, B_N);
        gather_comb_kernel<<<(n + TB - 1) / TB, TB, 0, stream>>>(item_i0, ie, g1_i, g2_i, add_w, xi, B_N);
    }

    // -------- filter MLP via fp32 WMMA --------
    {
        // GEMM1: [B,64] @ fw1^T[64,128] -> [B,128]
        int tiles1 = (B_N / 16) * (128 / 16);
        int blocks1 = (tiles1 * 32 + TB - 1) / TB;
        gemm_leaky_wmma<<<blocks1, TB, 0, stream>>>(xu, fw1, fb1, hu, B_N, 128, 64);
        gemm_leaky_wmma<<<blocks1, TB, 0, stream>>>(xi, fw1, fb1, hi, B_N, 128, 64);
        // GEMM2: [B,128] @ fw2^T[128,64] -> [B,64]
        int tiles2 = (B_N / 16) * (64 / 16);
        int blocks2 = (tiles2 * 32 + TB - 1) / TB;
        gemm_leaky_wmma<<<blocks2, TB, 0, stream>>>(hu, fw2, fb2, fu, B_N, 64, 128);
        gemm_leaky_wmma<<<blocks2, TB, 0, stream>>>(hi, fw2, fb2, fi, B_N, 64, 128);
    }

    // -------- prediction + MSE --------
    {
        int waves_per_block = TB / 32;
        int blocks = (B_N + waves_per_block - 1) / waves_per_block;
        pred_loss_kernel<<<blocks, TB, 0, stream>>>(fu, fi, user0, item_i0,
                                                    user_bias, item_bias, avg, ratings,
                                                    scal, B_N);
    }

    finalize_kernel<<<1, 64, 0, stream>>>(scal, out);
    (void)in_sizes; (void)n_in; (void)out_size; (void)ws_size;
}